// GaussianAdapter_50225347559934
// MI455X (gfx1250) — compile-verified
//
#include <hip/hip_runtime.h>
#include <hip/hip_bf16.h>
#include <math.h>

#define B_    4
#define HW_   16384
#define SPP_  3
#define C_    64
#define NG    (B_*HW_*SPP_)      /* 196608 gaussians */
#define DSH   25
#define RG_STRIDE 82             /* 3 + 4 + 75 */

/* workspace layout (floats):
   [0 .. 127]            per-batch header, 32 floats each:
                         [0]=multiplier, [1..9]=Kinv, [10..18]=Rc2w, [19..21]=origin
   [128 .. 128+4*896)    per-batch WMMA B operand DT[28][32]:
                         DT[k][n] = mask[k] * Dfull[n][k]  (zero padded)           */
#define HDR_STRIDE 32
#define DT_OFF     (B_*HDR_STRIDE)
#define DT_K       28
#define DT_N       32
#define DT_STRIDE  (DT_K*DT_N)

typedef float v2f __attribute__((ext_vector_type(2)));
typedef float v8f __attribute__((ext_vector_type(8)));

/* ---------------- small double-precision matrix helpers (n <= 9) ------------- */

__device__ static void matmul_d(const double* A, const double* Bm, double* Cm, int n) {
  for (int i = 0; i < n; i++)
    for (int j = 0; j < n; j++) {
      double s = 0.0;
      for (int k = 0; k < n; k++) s += A[i*n+k] * Bm[k*n+j];
      Cm[i*n+j] = s;
    }
}

/* expm via scaling-and-squaring + Taylor (double, n<=9) */
__device__ static void expm_d(double* M, int n, double* out) {
  double nrm = 0.0;
  for (int i = 0; i < n; i++) {
    double r = 0.0;
    for (int j = 0; j < n; j++) r += fabs(M[i*n+j]);
    nrm = fmax(nrm, r);
  }
  int s = 0;
  while (nrm > 0.5) { nrm *= 0.5; s++; }
  double sc = ldexp(1.0, -s);
  for (int i = 0; i < n*n; i++) M[i] *= sc;

  double term[81], tmp[81];
  for (int i = 0; i < n*n; i++) { out[i] = 0.0; term[i] = 0.0; }
  for (int i = 0; i < n; i++)   { out[i*n+i] = 1.0; term[i*n+i] = 1.0; }
  for (int k = 1; k <= 24; k++) {
    matmul_d(term, M, tmp, n);
    double inv = 1.0 / (double)k;
    for (int i = 0; i < n*n; i++) { term[i] = tmp[i]*inv; out[i] += term[i]; }
  }
  for (int q = 0; q < s; q++) {
    matmul_d(out, out, tmp, n);
    for (int i = 0; i < n*n; i++) out[i] = tmp[i];
  }
}

/* real SO(3) generators in the real-SH basis: X0 and X1.
   Faithful port of _su2_generators + _change_basis + Re(Q^H X Q); the (-1j)^l phase
   on Q cancels in Q^H X Q and is omitted. */
__device__ static void build_so3(int l, double* X0out, double* X1out) {
  const int n = 2*l + 1;
  const double jj = (double)l;
  double A0re[81], A1im[81];             /* X0 is real, X1 is purely imaginary */
  for (int i = 0; i < n*n; i++) { A0re[i] = 0.0; A1im[i] = 0.0; }
  for (int i = 0; i < n-1; i++) {
    double mr = -jj + (double)i;
    double rv = -sqrt(jj*(jj+1.0) - mr*(mr+1.0));   /* raising at (i+1,i) */
    double ml = -jj + 1.0 + (double)i;
    double lv =  sqrt(jj*(jj+1.0) - ml*(ml-1.0));   /* lowering at (i,i+1) */
    A0re[(i+1)*n + i] += 0.5*rv;
    A0re[i*n + (i+1)] += 0.5*lv;
  }
  for (int i = 0; i < n; i++) A1im[i*n+i] = -jj + (double)i;   /* diag(1j*m) */

  double Qre[81], Qim[81];
  for (int i = 0; i < n*n; i++) { Qre[i] = 0.0; Qim[i] = 0.0; }
  const double is2 = 0.70710678118654752440;
  for (int m = -l; m < 0; m++) {
    Qre[(l+m)*n + (l-m)] =  is2;    /* col l+|m| */
    Qim[(l+m)*n + (l+m)] = -is2;    /* col l-|m|, value -1j/sqrt2 */
  }
  Qre[l*n + l] = 1.0;
  for (int m = 1; m <= l; m++) {
    double sgn = (m & 1) ? -1.0 : 1.0;
    Qre[(l+m)*n + (l+m)] = sgn*is2;
    Qim[(l+m)*n + (l-m)] = sgn*is2; /* 1j*(-1)^m/sqrt2 */
  }

  double Tre[81], Tim[81];
  /* X0: A real.  T = A0*Q ; out = Re(Q^H T) */
  for (int i = 0; i < n; i++)
    for (int j = 0; j < n; j++) {
      double sr = 0.0, si = 0.0;
      for (int k = 0; k < n; k++) { double a = A0re[i*n+k]; sr += a*Qre[k*n+j]; si += a*Qim[k*n+j]; }
      Tre[i*n+j] = sr; Tim[i*n+j] = si;
    }
  for (int i = 0; i < n; i++)
    for (int j = 0; j < n; j++) {
      double s = 0.0;
      for (int k = 0; k < n; k++) s += Qre[k*n+i]*Tre[k*n+j] + Qim[k*n+i]*Tim[k*n+j];
      X0out[i*n+j] = s;
    }
  /* X1: A = i*D (pure imaginary).  T = A1*Q: Tre = -A1im*Qim, Tim = A1im*Qre */
  for (int i = 0; i < n; i++)
    for (int j = 0; j < n; j++) {
      double sr = 0.0, si = 0.0;
      for (int k = 0; k < n; k++) { double a = A1im[i*n+k]; sr -= a*Qim[k*n+j]; si += a*Qre[k*n+j]; }
      Tre[i*n+j] = sr; Tim[i*n+j] = si;
    }
  for (int i = 0; i < n; i++)
    for (int j = 0; j < n; j++) {
      double s = 0.0;
      for (int k = 0; k < n; k++) s += Qre[k*n+i]*Tre[k*n+j] + Qim[k*n+i]*Tim[k*n+j];
      X1out[i*n+j] = s;
    }
}

/* --------------------- kernel 1: per-batch precompute ------------------------ */

__global__ void precompute_kernel(const float* __restrict__ extr,
                                  const float* __restrict__ intr,
                                  const int* __restrict__ ih,
                                  const int* __restrict__ iw,
                                  float* __restrict__ ws) {
  const int t = threadIdx.x;
  /* zero-pad the WMMA B operand region (ws is poisoned by the harness) */
  for (int i = t; i < B_*DT_STRIDE; i += blockDim.x) ws[DT_OFF + i] = 0.0f;
  __syncthreads();

  if (t < B_) {
    const int b = t;
    const float* K = intr + b*9;
    /* multiplier = 0.1 * sum_i (K2inv @ pixel_size)_i */
    double a = K[0], bb = K[1], c2 = K[3], d2 = K[4];
    double det2 = a*d2 - bb*c2;
    double px = 1.0/(double)iw[0], py = 1.0/(double)ih[0];
    double i00 =  d2/det2, i01 = -bb/det2, i10 = -c2/det2, i11 = a/det2;
    double mult = 0.1 * ((i00*px + i01*py) + (i10*px + i11*py));
    /* Kinv via adjugate */
    double m00=K[0],m01=K[1],m02=K[2],m10=K[3],m11=K[4],m12=K[5],m20=K[6],m21=K[7],m22=K[8];
    double det = m00*(m11*m22-m12*m21) - m01*(m10*m22-m12*m20) + m02*(m10*m21-m11*m20);
    double id = 1.0/det;
    double kinv[9] = {
      (m11*m22-m12*m21)*id, (m02*m21-m01*m22)*id, (m01*m12-m02*m11)*id,
      (m12*m20-m10*m22)*id, (m00*m22-m02*m20)*id, (m02*m10-m00*m12)*id,
      (m10*m21-m11*m20)*id, (m01*m20-m00*m21)*id, (m00*m11-m01*m10)*id };
    float* hdr = ws + b*HDR_STRIDE;
    hdr[0] = (float)mult;
    for (int i = 0; i < 9; i++) hdr[1+i] = (float)kinv[i];
    const float* E = extr + b*16;
    for (int i = 0; i < 3; i++)
      for (int j = 0; j < 3; j++) hdr[10 + i*3 + j] = E[i*4 + j];
    hdr[19] = E[3]; hdr[20] = E[7]; hdr[21] = E[11];
  }

  if (t >= 8 && t < 8 + B_*5) {
    const int idx = t - 8;
    const int b = idx / 5, l = idx % 5;
    const float* E = extr + b*16;
    double R[9];
    for (int i = 0; i < 3; i++)
      for (int j = 0; j < 3; j++) R[i*3+j] = (double)E[i*4+j];
    /* _matrix_to_angles */
    double x0 = R[1], x1 = R[4], x2 = R[7];
    double nx = sqrt(x0*x0 + x1*x1 + x2*x2);
    x0 /= nx; x1 /= nx; x2 /= nx;
    double beta  = acos(fmin(1.0, fmax(-1.0, x1)));
    double alpha = atan2(x0, x2);
    double ca = cos(alpha), sa = sin(alpha), cb = cos(beta), sb = sin(beta);
    /* A = rotY(alpha) @ rotX(beta) */
    double Am[9] = { ca, sa*sb,  sa*cb,
                     0.0, cb,   -sb,
                    -sa, ca*sb,  ca*cb };
    /* Rp = A^T @ R ; need Rp[0][0], Rp[0][2] */
    double Rp00 = Am[0]*R[0] + Am[3]*R[3] + Am[6]*R[6];
    double Rp02 = Am[0]*R[2] + Am[3]*R[5] + Am[6]*R[8];
    double gamma = atan2(Rp02, Rp00);

    const int n = 2*l + 1;
    double X0[81], X1[81], M[81], Ea[81], Eb[81], Eg[81], T1[81], Ed[81];
    build_so3(l, X0, X1);
    for (int i = 0; i < n*n; i++) M[i] = alpha * X1[i];
    expm_d(M, n, Ea);
    for (int i = 0; i < n*n; i++) M[i] = -beta * X0[i];   /* _wigner_D(l, alpha, -beta, gamma) */
    expm_d(M, n, Eb);
    for (int i = 0; i < n*n; i++) M[i] = gamma * X1[i];
    expm_d(M, n, Eg);
    matmul_d(Ea, Eb, T1, n);
    matmul_d(T1, Eg, Ed, n);

    const double maskv[5] = {1.0, 0.025, 0.00625, 0.0015625, 0.000390625};
    const double mk = maskv[l];
    const int off = l*l;
    float* DT = ws + DT_OFF + b*DT_STRIDE;
    /* GEMM B operand: B[k][n] = mask[k] * Dfull[n][k] (k = input coeff, n = output) */
    for (int i = 0; i < n; i++)
      for (int j = 0; j < n; j++)
        DT[(off + j)*DT_N + (off + i)] = (float)(Ed[i*n + j] * mk);
  }
}

/* --------------------- kernel 2: per-gaussian pointwise ---------------------- */

__global__ void gaussian_main_kernel(const float* __restrict__ coords,
                                     const float* __restrict__ depths,
                                     const float* __restrict__ opac,
                                     const float* __restrict__ raw,
                                     const float* __restrict__ ws,
                                     float* __restrict__ out) {
  const int g = blockIdx.x*blockDim.x + threadIdx.x;
  if (g >= NG) return;
  const int b = g / (HW_*SPP_);
  const float* hdr  = ws + b*HDR_STRIDE;
  const float  mult = hdr[0];
  const float* kinv = hdr + 1;
  const float* Rw   = hdr + 10;
  const float* org  = hdr + 19;
  const float* rg   = raw + (size_t)g*RG_STRIDE;
  const float depth = depths[g];

  float sc[3];
#pragma unroll
  for (int i = 0; i < 3; i++) {
    float sg = 1.0f/(1.0f + __expf(-rg[i]));
    sc[i] = (0.5f + 14.5f*sg) * depth * mult;
  }
  float qr = rg[3], qi = rg[4], qj = rg[5], qk = rg[6];
  float inv = 1.0f/(sqrtf(qr*qr + qi*qi + qj*qj + qk*qk) + 1e-8f);
  qr *= inv; qi *= inv; qj *= inv; qk *= inv;
  float s2 = 2.0f/(qr*qr + qi*qi + qj*qj + qk*qk);
  float Rq[9];
  Rq[0] = 1.0f - s2*(qj*qj + qk*qk);
  Rq[1] = s2*(qi*qj - qk*qr);
  Rq[2] = s2*(qi*qk + qj*qr);
  Rq[3] = s2*(qi*qj + qk*qr);
  Rq[4] = 1.0f - s2*(qi*qi + qk*qk);
  Rq[5] = s2*(qj*qk - qi*qr);
  Rq[6] = s2*(qi*qk - qj*qr);
  Rq[7] = s2*(qj*qk + qi*qr);
  Rq[8] = 1.0f - s2*(qi*qi + qj*qj);
  const float sv[3] = {sc[0]*sc[0], sc[1]*sc[1], sc[2]*sc[2]};
  float cov[9];
#pragma unroll
  for (int i = 0; i < 3; i++)
#pragma unroll
    for (int k2 = 0; k2 < 3; k2++)
      cov[i*3+k2] = Rq[i*3+0]*sv[0]*Rq[k2*3+0]
                  + Rq[i*3+1]*sv[1]*Rq[k2*3+1]
                  + Rq[i*3+2]*sv[2]*Rq[k2*3+2];

  const float cx = coords[(size_t)g*2 + 0];
  const float cy = coords[(size_t)g*2 + 1];
  float v0 = kinv[0]*cx + kinv[1]*cy + kinv[2];
  float v1 = kinv[3]*cx + kinv[4]*cy + kinv[5];
  float v2 = kinv[6]*cx + kinv[7]*cy + kinv[8];
  const float dn = rsqrtf(v0*v0 + v1*v1 + v2*v2);
  v0 *= dn; v1 *= dn; v2 *= dn;
  const float w0 = Rw[0]*v0 + Rw[1]*v1 + Rw[2]*v2;
  const float w1 = Rw[3]*v0 + Rw[4]*v1 + Rw[5]*v2;
  const float w2 = Rw[6]*v0 + Rw[7]*v1 + Rw[8]*v2;

  const size_t N = (size_t)NG;
  float* mp = out + (size_t)g*3;
  mp[0] = org[0] + w0*depth;
  mp[1] = org[1] + w1*depth;
  mp[2] = org[2] + w2*depth;
  float* cp = out + 3*N + (size_t)g*9;
#pragma unroll
  for (int i = 0; i < 9; i++) cp[i] = cov[i];
  out[87*N + g] = opac[g];
  float* sp = out + 152*N + (size_t)g*3;
  sp[0] = sc[0]; sp[1] = sc[1]; sp[2] = sc[2];
  float* rp = out + 155*N + (size_t)g*4;
  rp[0] = qr; rp[1] = qi; rp[2] = qj; rp[3] = qk;
}

/* ------------- kernel 3: SH rotation as WMMA f32 16x16x4 GEMM --------------- */
/* Per batch: O[R,25] = V[R,25] x B[25,25], R = HW*SPP*3 rows (gaussian,channel).
   One wave handles 16 rows: 2 N-tiles x 7 K-chunks = 14 v_wmma_f32_16x16x4_f32.
   All 8 waves of a block share one batch (9216 tiles/batch, 8 tiles/block), so
   the B operand DT[28][32] is staged once per block into LDS. Output index is
   linearized: gib*75 + ch*25 + col == orow*25 + col (75 = 3*25) -> no div/3. */

#define TILES_PER_BATCH ((HW_*SPP_*3)/16)       /* 9216 */
#define BLOCKS_PER_BATCH (TILES_PER_BATCH/8)    /* 1152 */

__global__ void sh_rotate_wmma_kernel(const float* __restrict__ raw,
                                      const float* __restrict__ ws,
                                      float* __restrict__ out_sh) {
  __shared__ float sDT[DT_STRIDE];              /* 28x32 = 3.5 KB */
  const int b = blockIdx.x / BLOCKS_PER_BATCH;

  for (int i = threadIdx.x; i < DT_STRIDE; i += 256)
    sDT[i] = ws[DT_OFF + b*DT_STRIDE + i];
  __syncthreads();

  const int tile = (blockIdx.x - b*BLOCKS_PER_BATCH)*8 + (threadIdx.x >> 5);
  const int lane = threadIdx.x & 31;
  const int row0 = tile * 16;

  /* A 16x4 f32 layout: lane L -> M = L&15 ; VGPR0/1 hold K = koff / koff+1,
     koff = 0 (lanes 0-15) or 2 (lanes 16-31). */
  const int m    = lane & 15;
  const int koff = (lane >> 4) << 1;
  const int r    = row0 + m;
  const int gib  = r / 3;                       /* once per wave, unavoidable (stride 82) */
  const int ch   = r - gib*3;
  const float* shrow = raw + ((size_t)(b*HW_*SPP_) + gib)*RG_STRIDE + 7 + ch*DSH;

  v2f a[7];
#pragma unroll
  for (int kc = 0; kc < 7; kc++) {
    const int k0 = kc*4 + koff;
    a[kc].x = (k0     < DSH) ? shrow[k0]     : 0.0f;
    a[kc].y = (k0 + 1 < DSH) ? shrow[k0 + 1] : 0.0f;
  }

  const int nn    = lane & 15;
  const int mbase = (lane >> 4) << 3;           /* C/D: lanes 16-31 hold M+8 */
  float* outb = out_sh + (size_t)b*HW_*SPP_*75;
#pragma unroll
  for (int nt = 0; nt < 2; nt++) {
    v8f c = {0.f,0.f,0.f,0.f,0.f,0.f,0.f,0.f};
#pragma unroll
    for (int kc = 0; kc < 7; kc++) {
      const int k0 = kc*4 + koff;
      v2f bf;                     /* B 4x16: lane L -> N = L&15, rows koff/koff+1 */
      bf.x = sDT[k0*DT_N + nt*16 + nn];
      bf.y = sDT[(k0 + 1)*DT_N + nt*16 + nn];
      c = __builtin_amdgcn_wmma_f32_16x16x4_f32(
          false, a[kc], false, bf, (short)0, c, false, false);
    }
    const int col = nt*16 + nn;
    if (col < DSH) {
      const size_t base = (size_t)(row0 + mbase)*25 + col;
#pragma unroll
      for (int rr = 0; rr < 8; rr++)
        outb[base + (size_t)rr*25] = c[rr];
    }
  }
}

/* ------------- kernel 4: feature transpose + SPP broadcast ------------------ */

__global__ void feat_transpose_kernel(const float* __restrict__ grd,
                                      float* __restrict__ outf) {
  __shared__ float tile[C_][33];
  const int b   = blockIdx.y;
  const int hw0 = blockIdx.x * 32;
  const int t   = threadIdx.x;           /* 256 threads */
  const int hwi = t & 31;
  const int c0  = t >> 5;                /* 0..7 */
#pragma unroll
  for (int cc = c0; cc < C_; cc += 8)
    tile[cc][hwi] = grd[((size_t)b*C_ + cc)*HW_ + hw0 + hwi];  /* coalesced over hw */
  __syncthreads();
  const int c  = t & 63;
  const int r0 = t >> 6;                 /* 0..3 */
  for (int hh = r0; hh < 32; hh += 4) {
    const float v = tile[c][hh];
    const size_t base = ((size_t)b*HW_ + hw0 + hh)*(SPP_*C_) + c;  /* coalesced over c */
    outf[base]        = v;
    outf[base + C_]   = v;
    outf[base + 2*C_] = v;
  }
}

/* --------------------------------- launcher --------------------------------- */

extern "C" void kernel_launch(void* const* d_in, const int* in_sizes, int n_in,
                              void* d_out, int out_size, void* d_ws, size_t ws_size,
                              hipStream_t stream) {
  const float* extr   = (const float*)d_in[0];
  const float* intr   = (const float*)d_in[1];
  const float* coords = (const float*)d_in[2];
  const float* depths = (const float*)d_in[3];
  const float* opac   = (const float*)d_in[4];
  const float* raw    = (const float*)d_in[5];
  const float* grd    = (const float*)d_in[6];
  const int*   ih     = (const int*)d_in[7];
  const int*   iw     = (const int*)d_in[8];
  float* out = (float*)d_out;
  float* ws  = (float*)d_ws;

  precompute_kernel<<<1, 64, 0, stream>>>(extr, intr, ih, iw, ws);

  gaussian_main_kernel<<<NG/256, 256, 0, stream>>>(coords, depths, opac, raw, ws, out);

  sh_rotate_wmma_kernel<<<B_*BLOCKS_PER_BATCH, 256, 0, stream>>>(
      raw, ws, out + (size_t)12*NG);

  {
    dim3 gridF(HW_/32, B_);
    feat_transpose_kernel<<<gridF, 256, 0, stream>>>(grd, out + (size_t)88*NG);
  }
}